// res_gcn_d_38319698215331
// MI455X (gfx1250) — compile-verified
//
#include <hip/hip_runtime.h>

typedef __attribute__((ext_vector_type(2))) float v2f;
typedef __attribute__((ext_vector_type(8))) float v8f;

constexpr int B_ = 8;
constexpr int N_ = 4096;
constexpr int C_ = 128;
constexpr int K_ = 20;
constexpr float NEG_SLOPE = 0.01f;

__device__ __forceinline__ float leaky(float v) {
    return v > 0.0f ? v : NEG_SLOPE * v;
}

// ---------------------------------------------------------------------------
// Kernel 1: lpT[b][n][c] = leaky_relu(points[b][c][n])   (LDS tile transpose)
// ---------------------------------------------------------------------------
__global__ void __launch_bounds__(1024)
transpose_leaky_kernel(const float* __restrict__ points, float* __restrict__ lpT) {
    __shared__ float tile[32][33];
    const int b  = blockIdx.z;
    const int n0 = blockIdx.x * 32;
    const int c0 = blockIdx.y * 32;
    const int tx = threadIdx.x, ty = threadIdx.y;

    // coalesced read along n
    float v = points[(size_t)b * C_ * N_ + (size_t)(c0 + ty) * N_ + (n0 + tx)];
    tile[ty][tx] = leaky(v);
    __syncthreads();
    // coalesced write along c
    lpT[((size_t)b * N_ + (n0 + ty)) * C_ + (c0 + tx)] = tile[tx][ty];
}

// ---------------------------------------------------------------------------
// Kernel 2: brute-force kNN. One thread per (b,n) point. Maintains a sorted
// list of 21 packed keys (dist_bits<<32 | candidate_idx); distances are >= 0
// so fp32 bit patterns are monotone and the low 32 bits give a stable
// lowest-index tiebreak (matches jax top_k stability). Entry 0 is self.
// ---------------------------------------------------------------------------
__global__ void __launch_bounds__(256)
knn_kernel(const float* __restrict__ xyz, int* __restrict__ idxbuf) {
    const int r = blockIdx.x * blockDim.x + threadIdx.x;   // 0 .. B*N-1
    const int b = r >> 12;                                  // N_ == 4096
    const int n = r & (N_ - 1);
    const float* X = xyz + (size_t)b * 3 * N_;

    const float x = X[n], y = X[N_ + n], z = X[2 * N_ + n];

    unsigned long long list[K_ + 1];
#pragma unroll
    for (int i = 0; i <= K_; ++i) list[i] = ~0ull;

    for (int m = 0; m < N_; ++m) {
        const float dx = x - X[m];
        const float dy = y - X[N_ + m];
        const float dz = z - X[2 * N_ + m];
        const float d  = dx * dx + dy * dy + dz * dz;
        const unsigned long long key =
            ((unsigned long long)__float_as_uint(d) << 32) | (unsigned int)m;
        if (key < list[K_]) {
            int j = K_;
            while (j > 0 && list[j - 1] > key) {
                list[j] = list[j - 1];
                --j;
            }
            list[j] = key;
        }
    }
    // drop entry 0 (self), write 20 neighbor indices
#pragma unroll
    for (int k = 1; k <= K_; ++k)
        idxbuf[(size_t)r * K_ + (k - 1)] = (int)(list[k] & 0xffffffffu);
}

// ---------------------------------------------------------------------------
// Kernel 3: gsumT[b][n][c] = sum_k lpT[b][ idx[b][n][k] ][c]
// One block per point, 128 lanes over channels -> 512B coalesced gathers.
// ---------------------------------------------------------------------------
__global__ void __launch_bounds__(128)
gather_sum_kernel(const float* __restrict__ lpT, const int* __restrict__ idxbuf,
                  float* __restrict__ gsumT) {
    const int r = blockIdx.x;          // b*N + n
    const int c = threadIdx.x;         // 0..127
    const int b = r >> 12;

    float s = 0.0f;
#pragma unroll 4
    for (int k = 0; k < K_; ++k) {
        const int m = idxbuf[(size_t)r * K_ + k];        // uniform across block
        s += lpT[((size_t)b * N_ + m) * C_ + c];         // coalesced row
    }
    gsumT[(size_t)r * C_ + c] = s;
}

// ---------------------------------------------------------------------------
// Kernel 4: fused double-GEMM + epilogue using V_WMMA_F32_16X16X4_F32.
//   out[b][o][n] = ((Wc @ leaky(P))[o][n] + (Wg @ G)[o][n]
//                   + bc[o] + K*bg[o]) / (K+1) + P[o][n]
// One wave per 16(o) x 64(n) output tile; A tile (weights) reused across the
// four 16-wide n subtiles. K-loop: 32 steps over Wc/P + 32 steps over Wg/G.
//
// f32 WMMA operand layout (ISA 7.12.2):
//   A 16x4 : lane<16 -> M=lane, regs {K=0,K=1}; lane>=16 -> same M, {K=2,K=3}
//   B 4x16 : lane<16 -> N=lane, regs {K=0,K=1}; lane>=16 -> same N, {K=2,K=3}
//   C/D    : reg v, lane l -> M = v + (l<16 ? 0 : 8), N = l&15
// ---------------------------------------------------------------------------
__global__ void __launch_bounds__(32)
fused_gemm_kernel(const float* __restrict__ points, const float* __restrict__ Wc,
                  const float* __restrict__ bc, const float* __restrict__ Wg,
                  const float* __restrict__ bg, const float* __restrict__ gsumT,
                  float* __restrict__ out) {
    const int lane = threadIdx.x;
    const int b    = blockIdx.z;
    const int o0   = blockIdx.y * 16;
    const int n0   = blockIdx.x * 64;
    const int half = lane >> 4;          // 0 or 1
    const int l15  = lane & 15;
    const int koff = half * 2;           // K offset carried by this lane half

    const float* Pb = points + (size_t)b * C_ * N_;
    const float* Gb = gsumT + (size_t)b * N_ * C_;

    v8f acc[4] = {};

    // Phase 1: Wc @ leaky(points)  (B operand: column n of P, strided rows)
    for (int kk = 0; kk < C_; kk += 4) {
        v2f a;
        a.x = Wc[(size_t)(o0 + l15) * C_ + (kk + koff)];
        a.y = Wc[(size_t)(o0 + l15) * C_ + (kk + koff + 1)];
#pragma unroll
        for (int t = 0; t < 4; ++t) {
            const int nn = n0 + t * 16 + l15;
            v2f bb;
            bb.x = leaky(Pb[(size_t)(kk + koff) * N_ + nn]);
            bb.y = leaky(Pb[(size_t)(kk + koff + 1) * N_ + nn]);
            acc[t] = __builtin_amdgcn_wmma_f32_16x16x4_f32(
                false, a, false, bb, (short)0, acc[t], false, false);
        }
    }

    // Phase 2: Wg @ gsum  (B operand: row n of gsumT, contiguous channels)
    for (int kk = 0; kk < C_; kk += 4) {
        v2f a;
        a.x = Wg[(size_t)(o0 + l15) * C_ + (kk + koff)];
        a.y = Wg[(size_t)(o0 + l15) * C_ + (kk + koff + 1)];
#pragma unroll
        for (int t = 0; t < 4; ++t) {
            const int nn = n0 + t * 16 + l15;
            v2f bb;
            bb.x = Gb[(size_t)nn * C_ + (kk + koff)];
            bb.y = Gb[(size_t)nn * C_ + (kk + koff + 1)];
            acc[t] = __builtin_amdgcn_wmma_f32_16x16x4_f32(
                false, a, false, bb, (short)0, acc[t], false, false);
        }
    }

    // Epilogue: bias, pool-average, residual add
    const float inv = 1.0f / (float)(K_ + 1);
#pragma unroll
    for (int t = 0; t < 4; ++t) {
        const int nn = n0 + t * 16 + l15;
#pragma unroll
        for (int v = 0; v < 8; ++v) {
            const int o = o0 + v + half * 8;
            const float bias = bc[o] + (float)K_ * bg[o];
            const float r = (acc[t][v] + bias) * inv + Pb[(size_t)o * N_ + nn];
            out[((size_t)b * C_ + o) * N_ + nn] = r;
        }
    }
}

// ---------------------------------------------------------------------------
// Launch
// ---------------------------------------------------------------------------
extern "C" void kernel_launch(void* const* d_in, const int* in_sizes, int n_in,
                              void* d_out, int out_size, void* d_ws, size_t ws_size,
                              hipStream_t stream) {
    const float* xyz    = (const float*)d_in[0];
    const float* points = (const float*)d_in[1];
    const float* Wc     = (const float*)d_in[2];
    const float* bc     = (const float*)d_in[3];
    const float* Wg     = (const float*)d_in[4];
    const float* bg     = (const float*)d_in[5];
    float* out = (float*)d_out;

    // workspace layout
    char* ws = (char*)d_ws;
    const size_t lpT_bytes = (size_t)B_ * N_ * C_ * sizeof(float);   // 16 MB
    const size_t idx_bytes = (size_t)B_ * N_ * K_ * sizeof(int);     // 2.5 MB
    float* lpT    = (float*)ws;
    int*   idxbuf = (int*)(ws + lpT_bytes);
    float* gsumT  = (float*)(ws + lpT_bytes + idx_bytes);            // 16 MB

    transpose_leaky_kernel<<<dim3(N_ / 32, C_ / 32, B_), dim3(32, 32), 0, stream>>>(
        points, lpT);

    knn_kernel<<<(B_ * N_) / 256, 256, 0, stream>>>(xyz, idxbuf);

    gather_sum_kernel<<<B_ * N_, C_, 0, stream>>>(lpT, idxbuf, gsumT);

    fused_gemm_kernel<<<dim3(N_ / 64, C_ / 16, B_), 32, 0, stream>>>(
        points, Wc, bc, Wg, bg, gsumT, out);
}